// GaussianSplattingRenderer_57750130262479
// MI455X (gfx1250) — compile-verified
//
#include <hip/hip_runtime.h>
#include <hip/hip_bf16.h>
#include <math.h>

#define TILE 16
#define MAXG 1024

// Packed per-gaussian record: 12 floats = 48 bytes, 16B-aligned for B128 copies.
struct alignas(16) GP {
    float u, v, ia, ibc;     // center + inverse-conic (ia, ib+ic, idd)
    float idd, op, z, rad;   // opacity, camera depth, 3-sigma pixel radius
    float cr, cg, cb, pad;   // sigmoid color
};

// -------- CDNA5 async global->LDS staging (gfx1250) --------
#if __has_builtin(__builtin_amdgcn_global_load_async_to_lds_b128)
#define HAVE_ASYNC_LDS 1
#else
#define HAVE_ASYNC_LDS 0
#if defined(__AMDGCN__)
#warning "gfx1250 async-to-LDS builtin not found on device pass; using direct LDS copy fallback"
#endif
#endif

// Builtin wants int __vector(4) pointers: global src in AS(1), LDS dst in AS(3).
typedef int v4i __attribute__((ext_vector_type(4)));
typedef __attribute__((address_space(1))) v4i* gv4_ptr;
typedef __attribute__((address_space(3))) v4i* lv4_ptr;

// ---------------- Kernel: per-pixel sequential blend ----------------
// (Placed first in the file so the disasm snippet shows the async staging.)
__global__ void __launch_bounds__(TILE*TILE)
gs_render(const float4* __restrict__ gp4, float* __restrict__ out,
          int n, int W, int H)
{
    __shared__ GP sg[MAXG];                       // 48 KB resident table
    float4* s4 = reinterpret_cast<float4*>(sg);

    int tid  = threadIdx.y * blockDim.x + threadIdx.x;
    int nvec = n * 3;

#if HAVE_ASYNC_LDS
    // gfx1250 async copies: tracked by ASYNCcnt, no VGPR return traffic.
    // 12 x 4 KB per issue round (256 lanes x b128) covers the 48 KB table.
#pragma unroll
    for (int k = 0; k < (MAXG * 3) / (TILE * TILE); ++k) {
        int i = tid + k * (TILE * TILE);
        if (i < nvec) {
            __builtin_amdgcn_global_load_async_to_lds_b128(
                (gv4_ptr)(gp4 + i), (lv4_ptr)(s4 + i), 0, 0);
        }
    }
#if __has_builtin(__builtin_amdgcn_s_wait_asynccnt)
    __builtin_amdgcn_s_wait_asynccnt(0);
#else
    asm volatile("s_wait_asynccnt 0x0" ::: "memory");
#endif
#else
    for (int i = tid; i < nvec; i += TILE*TILE) s4[i] = gp4[i];
#endif
    __syncthreads();

    int x = blockIdx.x * TILE + threadIdx.x;
    int y = blockIdx.y * TILE + threadIdx.y;
    float px = (float)x, py = (float)y;

    float ir = 0.0f, ig = 0.0f, ib = 0.0f;
    float abuf  = 0.0f;
    float depth = __builtin_inff();

    // Sequential over gaussians in index order (matches lax.scan semantics).
    // x2 unroll: batches the broadcast ds_load_b128s and lets v_exp_f32
    // (TRANS pipe) co-execute with the other gaussian's blend VALU.
#pragma unroll 2
    for (int g = 0; g < n; ++g) {
        GP p = sg[g];                              // broadcast ds_load_b128 x3
        float dx = px - p.u;
        float dy = py - p.v;
        float ds = p.ia*dx*dx + p.ibc*dx*dy + p.idd*dy*dy;

        float min_u = fmaxf(0.0f, truncf(p.u - p.rad));
        float max_u = truncf(p.u + p.rad);
        float min_v = fmaxf(0.0f, truncf(p.v - p.rad));
        float max_v = truncf(p.v + p.rad);
        bool inbox = (px >= min_u) && (px <= max_u) &&
                     (py >= min_v) && (py <= max_v);
        bool mask  = inbox && (ds < 9.0f) && (p.z < depth);

        float alpha = p.op * __expf(-0.5f * ds);
        float na = mask ? alpha * (1.0f - abuf) : 0.0f;
        float om = 1.0f - na;
        ir = ir*om + p.cr*na;
        ig = ig*om + p.cg*na;
        ib = ib*om + p.cb*na;
        abuf += na;
        if (mask) depth = p.z;
    }

    if (x < W && y < H) {
        int o = (y * W + x) * 3;
        out[o + 0] = ir;
        out[o + 1] = ig;
        out[o + 2] = ib;
    }
}

// ---------------- Kernel: per-gaussian preprocessing ----------------
__global__ void gs_preprocess(const float* __restrict__ means,
                              const float* __restrict__ scales,
                              const float* __restrict__ rots,
                              const float* __restrict__ opac,
                              const float* __restrict__ feats,
                              const float* __restrict__ pose,
                              const float* __restrict__ pfocal,
                              const float* __restrict__ pcx,
                              const float* __restrict__ pcy,
                              float4* __restrict__ out4, int n)
{
    int i = blockIdx.x * blockDim.x + threadIdx.x;
    if (i >= n) return;

    float f = *pfocal, cx = *pcx, cy = *pcy;

    // camera pose (row-major 4x4): R rows + translation
    float R00 = pose[0], R01 = pose[1], R02 = pose[2],  t0 = pose[3];
    float R10 = pose[4], R11 = pose[5], R12 = pose[6],  t1 = pose[7];
    float R20 = pose[8], R21 = pose[9], R22 = pose[10], t2 = pose[11];

    float mx = means[3*i], my = means[3*i+1], mz = means[3*i+2];
    float s0 = __expf(scales[3*i]);
    float s1 = __expf(scales[3*i+1]);
    float s2 = __expf(scales[3*i+2]);

    // quaternion -> rotation (matches reference quat_to_rot)
    float qw = rots[4*i], qx = rots[4*i+1], qy = rots[4*i+2], qz = rots[4*i+3];
    float two_s = 2.0f / (qw*qw + qx*qx + qy*qy + qz*qz);
    float xx = qx*qx*two_s, xy = qx*qy*two_s, xz = qx*qz*two_s;
    float yw = qy*qw*two_s, yy = qy*qy*two_s, yz = qy*qz*two_s;
    float zw = qz*qw*two_s, zz = qz*qz*two_s, xw = qx*qw*two_s;
    float M00 = 1.f-(yy+zz), M01 = xy-zw,      M02 = xz+yw;
    float M10 = xy+zw,       M11 = 1.f-(xx+zz), M12 = yz-xw;
    float M20 = xz-yw,       M21 = yz+xw,       M22 = 1.f-(xx+yy);

    // camera-space mean, projection
    float mc0 = mx*R00 + my*R01 + mz*R02 + t0;
    float mc1 = mx*R10 + my*R11 + mz*R12 + t1;
    float mc2 = mx*R20 + my*R21 + mz*R22 + t2;
    float z  = mc2;
    float iz = 1.0f / z;
    float u  = mc0 * iz * f + cx;
    float v  = mc1 * iz * f + cy;

    // J = [[f,0,-cx],[0,f,-cy],[0,0,1]] @ R   (rows 0,1 only needed)
    float J00 = f*R00 - cx*R20, J01 = f*R01 - cx*R21, J02 = f*R02 - cx*R22;
    float J10 = f*R10 - cy*R20, J11 = f*R11 - cy*R21, J12 = f*R12 - cy*R22;

    // V[i][k] = (J @ M)[i][k] * sc[k]
    float V00 = (J00*M00 + J01*M10 + J02*M20) * s0;
    float V01 = (J00*M01 + J01*M11 + J02*M21) * s1;
    float V02 = (J00*M02 + J01*M12 + J02*M22) * s2;
    float V10 = (J10*M00 + J11*M10 + J12*M20) * s0;
    float V11 = (J10*M01 + J11*M11 + J12*M21) * s1;
    float V12 = (J10*M02 + J11*M12 + J12*M22) * s2;

    float iz2 = iz * iz;
    float a = (V00*V00 + V01*V01 + V02*V02) * iz2;
    float b = (V00*V10 + V01*V11 + V02*V12) * iz2;
    float d = (V10*V10 + V11*V11 + V12*V12) * iz2;
    float det  = a*d - b*b;
    float idet = 1.0f / det;
    float ia  =  d * idet;
    float ibc = -2.0f * b * idet;   // (ib + ic) pre-combined
    float idd =  a * idet;

    float smax = fmaxf(s0, fmaxf(s1, s2));
    float rad  = smax * f * iz * 3.0f;

    float op = 1.0f / (1.0f + __expf(-opac[i]));
    float cr = 1.0f / (1.0f + __expf(-feats[3*i]));
    float cg = 1.0f / (1.0f + __expf(-feats[3*i+1]));
    float cb = 1.0f / (1.0f + __expf(-feats[3*i+2]));

    out4[3*i + 0] = make_float4(u,   v,  ia, ibc);
    out4[3*i + 1] = make_float4(idd, op, z,  rad);
    out4[3*i + 2] = make_float4(cr,  cg, cb, 0.0f);
}

extern "C" void kernel_launch(void* const* d_in, const int* in_sizes, int n_in,
                              void* d_out, int out_size, void* d_ws, size_t ws_size,
                              hipStream_t stream)
{
    const float* means  = (const float*)d_in[0];
    const float* scales = (const float*)d_in[1];
    const float* rots   = (const float*)d_in[2];
    const float* opac   = (const float*)d_in[3];
    const float* feats  = (const float*)d_in[4];
    const float* pose   = (const float*)d_in[5];
    const float* focal  = (const float*)d_in[6];
    const float* cx     = (const float*)d_in[7];
    const float* cy     = (const float*)d_in[8];

    int n = in_sizes[0] / 3;           // N = 1024
    if (n > MAXG) n = MAXG;
    const int W = 128, H = 128;        // fixed by the reference constants

    float4* gp4 = (float4*)d_ws;       // 1024 * 48 B = 48 KB of scratch
    float*  img = (float*)d_out;

    gs_preprocess<<<(n + 255) / 256, 256, 0, stream>>>(
        means, scales, rots, opac, feats, pose, focal, cx, cy, gp4, n);

    dim3 grid((W + TILE - 1) / TILE, (H + TILE - 1) / TILE);
    dim3 block(TILE, TILE);
    gs_render<<<grid, block, 0, stream>>>(gp4, img, n, W, H);
}